// WhisperCrossAttentionStepTVM_24962349925059
// MI455X (gfx1250) — compile-verified
//
#include <hip/hip_runtime.h>
#include <math.h>

// ---------------------------------------------------------------------------
// WhisperCrossAttentionStep on gfx1250 (MI455X)
//   B=32, Q=1, SKV=1500, E=1280, H=20, Dh=64, all fp32.
//   Memory-bound: ~492MB of K/V streamed once -> ~21us floor @ 23.3 TB/s.
//   Projections (32x1280x1280) via V_WMMA_F32_16X16X4_F32.
//   K/V streamed with non-temporal b128 loads (TH_NT) to avoid L2 churn.
// ---------------------------------------------------------------------------

typedef __attribute__((ext_vector_type(2))) float v2f;
typedef __attribute__((ext_vector_type(4))) float v4f;
typedef __attribute__((ext_vector_type(8))) float v8f;

#define EMBED 1280
#define SKV   1500
#define NBH   640   // 32 batches * 20 heads
#define DH    64

// ---------------------------------------------------------------------------
// C[m,n] = (sum_k A[m,k] * W[n,k] + bias[n]) * scale
// A: [32, 1280] row-major, W: [1280, 1280] row-major (torch Linear weight)
// One wave (32 threads) per 16x16 output tile; grid = (Mtiles=2, Ntiles=80).
// K-loop: 320 x v_wmma_f32_16x16x4_f32 accumulate.
// ---------------------------------------------------------------------------
__global__ __launch_bounds__(32)
void proj_gemm_wmma(const float* __restrict__ A,
                    const float* __restrict__ W,
                    const float* __restrict__ bias,
                    float* __restrict__ C,
                    float scale)
{
    const int K = EMBED;
    const int mtile = blockIdx.x;   // 0..1
    const int ntile = blockIdx.y;   // 0..79
    const int lane  = threadIdx.x;  // 0..31
    const int half  = lane >> 4;    // selects K pair {0,1} vs {2,3}
    const int l16   = lane & 15;

    // A fragment (16x4 f32): lanes 0-15 hold rows M=0..15; VGPR0=K0/K2, VGPR1=K1/K3
    const float* Arow = A + (size_t)(mtile * 16 + l16) * K + 2 * half;
    // B fragment (4x16 f32): B[k][n] = W[n][k]; lane n reads its weight row
    const float* Wrow = W + (size_t)(ntile * 16 + l16) * K + 2 * half;

    v8f acc = {};
    #pragma unroll 4
    for (int k = 0; k < K; k += 4) {
        v2f a = *(const v2f*)(Arow + k);
        v2f b = *(const v2f*)(Wrow + k);
        acc = __builtin_amdgcn_wmma_f32_16x16x4_f32(
            /*neg_a=*/false, a, /*neg_b=*/false, b,
            /*c_mod=*/(short)0, acc, /*reuse_a=*/false, /*reuse_b=*/false);
    }

    // D layout: lane l16 = column n; half selects rows M=0..7 / M=8..15 (VGPR r -> M=r)
    const int n = ntile * 16 + l16;
    const float bv = bias[n];
    const int mbase = mtile * 16 + half * 8;
    #pragma unroll
    for (int r = 0; r < 8; ++r) {
        C[(size_t)(mbase + r) * EMBED + n] = (acc[r] + bv) * scale;
    }
}

// ---------------------------------------------------------------------------
// Attention for one (b,h): scores = K@q (q pre-scaled), softmax, out = p@V.
// 256 threads (8 wave32). K and V each streamed once with NT b128 loads.
// ---------------------------------------------------------------------------
__global__ __launch_bounds__(256)
void cross_attn_kernel(const float* __restrict__ Q,   // [640, 64] pre-scaled
                       const float* __restrict__ Km,  // [640, 1500, 64]
                       const float* __restrict__ Vm,  // [640, 1500, 64]
                       float* __restrict__ O)         // [640, 64]
{
    const int bh  = blockIdx.x;
    const int tid = threadIdx.x;
    const int grp = tid >> 4;   // 0..15 : which kv-row in the group of 16
    const int ln  = tid & 15;   // 0..15 : which float4 of the 64-wide row

    __shared__ float s_sc[SKV];
    __shared__ float s_red[256];
    __shared__ float s_q[DH];
    __shared__ v4f   s_pacc[16][16];

    if (tid < DH) s_q[tid] = Q[(size_t)bh * DH + tid];
    __syncthreads();

    const float* Kb = Km + (size_t)bh * SKV * DH;
    const float* Vb = Vm + (size_t)bh * SKV * DH;
    const v4f qv = *(const v4f*)(s_q + ln * 4);

    // ---- Pass 1: scores[k] = dot(q, K[k,:]) ----
    for (int k = grp; k < SKV; k += 16) {
        __builtin_prefetch(Kb + (size_t)(k + 16) * DH + ln * 4, 0, 0);
        const v4f kv =
            __builtin_nontemporal_load((const v4f*)(Kb + (size_t)k * DH + ln * 4));
        float p = kv.x * qv.x + kv.y * qv.y + kv.z * qv.z + kv.w * qv.w;
        // reduce across the 16 lanes covering one row (within a wave32)
        p += __shfl_xor(p, 8, 32);
        p += __shfl_xor(p, 4, 32);
        p += __shfl_xor(p, 2, 32);
        p += __shfl_xor(p, 1, 32);
        if (ln == 0) s_sc[k] = p;
    }
    __syncthreads();

    // ---- Softmax: block max ----
    float m = -INFINITY;
    for (int k = tid; k < SKV; k += 256) m = fmaxf(m, s_sc[k]);
    s_red[tid] = m;
    __syncthreads();
    for (int s = 128; s > 0; s >>= 1) {
        if (tid < s) s_red[tid] = fmaxf(s_red[tid], s_red[tid + s]);
        __syncthreads();
    }
    const float gmax = s_red[0];
    __syncthreads();

    // ---- Softmax: exp + sum ----
    float sum = 0.f;
    for (int k = tid; k < SKV; k += 256) {
        const float e = __expf(s_sc[k] - gmax);
        s_sc[k] = e;
        sum += e;
    }
    s_red[tid] = sum;
    __syncthreads();
    for (int s = 128; s > 0; s >>= 1) {
        if (tid < s) s_red[tid] += s_red[tid + s];
        __syncthreads();
    }
    const float inv = 1.0f / s_red[0];
    __syncthreads();

    // ---- Pass 2: out[d] = sum_k p[k] * V[k,d] ----
    v4f acc = {};
    for (int k = grp; k < SKV; k += 16) {
        __builtin_prefetch(Vb + (size_t)(k + 16) * DH + ln * 4, 0, 0);
        const float p = s_sc[k];
        const v4f vv =
            __builtin_nontemporal_load((const v4f*)(Vb + (size_t)k * DH + ln * 4));
        acc.x += p * vv.x; acc.y += p * vv.y; acc.z += p * vv.z; acc.w += p * vv.w;
    }
    s_pacc[grp][ln] = acc;
    __syncthreads();

    if (tid < 16) {
        v4f a = s_pacc[0][tid];
        #pragma unroll
        for (int r = 1; r < 16; ++r) {
            const v4f b = s_pacc[r][tid];
            a.x += b.x; a.y += b.y; a.z += b.z; a.w += b.w;
        }
        v4f* o4 = (v4f*)(O + (size_t)bh * DH);
        v4f res; res.x = a.x * inv; res.y = a.y * inv; res.z = a.z * inv; res.w = a.w * inv;
        o4[tid] = res;
    }
}

// ---------------------------------------------------------------------------
extern "C" void kernel_launch(void* const* d_in, const int* in_sizes, int n_in,
                              void* d_out, int out_size, void* d_ws, size_t ws_size,
                              hipStream_t stream) {
    const float* hidden = (const float*)d_in[0];  // [32, 1, 1280]
    const float* ck     = (const float*)d_in[1];  // [32, 20, 1500, 64]
    const float* cv     = (const float*)d_in[2];  // [32, 20, 1500, 64]
    const float* qw     = (const float*)d_in[3];  // [1280, 1280]
    const float* qb     = (const float*)d_in[4];  // [1280]
    const float* ow     = (const float*)d_in[5];  // [1280, 1280]
    const float* ob     = (const float*)d_in[6];  // [1280]
    float* out = (float*)d_out;                   // [32, 1280]

    float* q_ws    = (float*)d_ws;                // [32, 1280] == [640, 64]
    float* attn_ws = q_ws + 32 * EMBED;           // [32, 1280] == [640, 64]

    const float scale = 0.125f;                   // 64^-0.5

    dim3 pgrid(2, 80);
    // q = (h @ q_w^T + q_b) * scale   (scale applies to bias too, per reference)
    proj_gemm_wmma<<<pgrid, 32, 0, stream>>>(hidden, qw, qb, q_ws, scale);
    // per-(b,h) softmax attention over 1500 kv
    cross_attn_kernel<<<NBH, 256, 0, stream>>>(q_ws, ck, cv, attn_ws);
    // out = attn @ out_w^T + out_b
    proj_gemm_wmma<<<pgrid, 32, 0, stream>>>(attn_ws, ow, ob, out, 1.0f);
}